// m2gate_55679956025638
// MI455X (gfx1250) — compile-verified
//
#include <hip/hip_runtime.h>
#include <hip/hip_bf16.h>

// Problem constants (from reference)
#define BB   32      // batch
#define TT   2048    // time
#define DD   1024    // feature dim
#define MEM  64      // memory slots
#define HH   512     // D/2
#define SPLIT 16     // T-chunks for mean reduction
#define CHUNK (TT / SPLIT)

typedef float v2f __attribute__((ext_vector_type(2)));
typedef float v8f __attribute__((ext_vector_type(8)));

// ---------------------------------------------------------------------------
// Kernel 1: partial sums over T chunks.  grid (B, SPLIT), block 256.
// Each thread owns one float4 column group -> fully coalesced B128 loads.
// 256 MiB streamed; this kernel is the entire roofline cost (~11us @ 23.3TB/s).
// ---------------------------------------------------------------------------
__global__ void mean_partial_kernel(const float* __restrict__ x,
                                    float* __restrict__ part) {
    const int b   = blockIdx.x;       // 0..31
    const int s   = blockIdx.y;       // 0..15
    const int tid = threadIdx.x;      // 0..255  (D/4 = 256 float4 per row)
    const float4* __restrict__ x4 =
        ((const float4*)x) + (size_t)b * TT * (DD / 4)
                           + (size_t)s * CHUNK * (DD / 4) + tid;
    float4 acc = make_float4(0.f, 0.f, 0.f, 0.f);
#pragma unroll 4
    for (int t = 0; t < CHUNK; ++t) {
        float4 v = x4[(size_t)t * (DD / 4)];
        acc.x += v.x; acc.y += v.y; acc.z += v.z; acc.w += v.w;
    }
    ((float4*)part)[((size_t)b * SPLIT + s) * (DD / 4) + tid] = acc;
}

// ---------------------------------------------------------------------------
// Kernel 2: combine partials -> mean (deterministic tree).  grid B, block 256.
// ---------------------------------------------------------------------------
__global__ void mean_final_kernel(const float* __restrict__ part,
                                  float* __restrict__ mean) {
    const int b   = blockIdx.x;
    const int tid = threadIdx.x;
    const float4* __restrict__ p =
        ((const float4*)part) + (size_t)b * SPLIT * (DD / 4) + tid;
    float4 acc = make_float4(0.f, 0.f, 0.f, 0.f);
#pragma unroll
    for (int s = 0; s < SPLIT; ++s) {
        float4 v = p[(size_t)s * (DD / 4)];
        acc.x += v.x; acc.y += v.y; acc.z += v.z; acc.w += v.w;
    }
    const float inv = 1.0f / (float)TT;
    acc.x *= inv; acc.y *= inv; acc.z *= inv; acc.w *= inv;
    ((float4*)mean)[(size_t)b * (DD / 4) + tid] = acc;
}

// ---------------------------------------------------------------------------
// Kernel 3: generic tiny-GEMM via V_WMMA_F32_16X16X4_F32 (wave32, f32 WMMA).
//   out[M=32, N] = act( A[32,K] @ W[N,K]^T + bias )
// One wave (32 threads) per 16x16 output tile.  grid = (M/16, N/16).
//
// Operand layouts per CDNA5 ISA §7.12.2 (32-bit A 16x4 / B 4x16):
//   A: lane l (l&15 = row M), VGPR0 = K(base)+2*(l>>4), VGPR1 = +1  -> float2
//   B: lane l (l&15 = col N), same K striping (B[k][n] = W[n][k])   -> float2
//   C/D: VGPR g -> row = g + 8*(l>>4), col = l&15
// ---------------------------------------------------------------------------
__global__ void wmma_gemm_f32_kernel(const float* __restrict__ A,
                                     const float* __restrict__ W,
                                     const float* __restrict__ bias,
                                     float* __restrict__ out,
                                     int N, int K, int act /*0=none,1=silu*/) {
    const int tileM = blockIdx.x * 16;
    const int tileN = blockIdx.y * 16;
    const int lane  = threadIdx.x;      // 0..31, one wave, EXEC all ones
    const int half  = lane >> 4;        // 0 or 1
    const int l15   = lane & 15;

    const float2* __restrict__ A2 = (const float2*)(A + (size_t)(tileM + l15) * K);
    const float2* __restrict__ W2 = (const float2*)(W + (size_t)(tileN + l15) * K);

    v8f c = {0.f, 0.f, 0.f, 0.f, 0.f, 0.f, 0.f, 0.f};

    for (int kk = 0; kk < K; kk += 4) {           // uniform scalar loop
        const int k2 = (kk >> 1) + half;          // float2 index
        float2 av = A2[k2];
        float2 bv = W2[k2];
        v2f a; a.x = av.x; a.y = av.y;
        v2f b; b.x = bv.x; b.y = bv.y;
        // D = A x B + C, fp32 WMMA (16x16x4)
        c = __builtin_amdgcn_wmma_f32_16x16x4_f32(
                /*neg_a=*/false, a, /*neg_b=*/false, b,
                /*c_mod=*/(short)0, c, /*reuse_a=*/false, /*reuse_b=*/false);
    }

    const int   n  = tileN + l15;
    const float bn = bias ? bias[n] : 0.0f;
#pragma unroll
    for (int g = 0; g < 8; ++g) {
        const int m = tileM + g + half * 8;
        float v = c[g] + bn;
        if (act == 1) v = v / (1.0f + __expf(-v));   // SiLU: x*sigmoid(x)
        out[(size_t)m * N + n] = v;
    }
}

// ---------------------------------------------------------------------------
// Wave reductions (wave32)
// ---------------------------------------------------------------------------
__device__ inline float wave_sum(float v) {
#pragma unroll
    for (int m = 16; m > 0; m >>= 1) v += __shfl_xor(v, m, 32);
    return v;
}
__device__ inline float wave_max(float v) {
#pragma unroll
    for (int m = 16; m > 0; m >>= 1) v = fmaxf(v, __shfl_xor(v, m, 32));
    return v;
}

// ---------------------------------------------------------------------------
// Kernel 4: everything small, fused.  1 block x 1024 threads (32 waves).
//  - xp row norms, key row norms
//  - cosine sims vs mkey0/1, argmax, change detector -> scalar output
//  - softmax over MEM=64 (normalized dots / sqrt(D)) -> mem_val
//  - mlp_out = h @ mlp_w2^T + b2 ;  gate = concat @ concat_w^T + cb
// ---------------------------------------------------------------------------
__global__ void finalize_kernel(const float* __restrict__ xp,     // [32,1024]
                                const float* __restrict__ dotk,   // [32,64] raw xp.key
                                const float* __restrict__ h,      // [32,512]
                                const float* __restrict__ mkey0,
                                const float* __restrict__ mkey1,
                                const float* __restrict__ val_matrix, // [64]
                                const float* __restrict__ mlp_w2,     // [512]
                                const float* __restrict__ mlp_b2,     // [1]
                                const float* __restrict__ concat_w,   // [2]
                                const float* __restrict__ concat_b,   // [1]
                                const float* __restrict__ key_matrix, // [64,1024]
                                const float* __restrict__ threshold,  // [1]
                                float* __restrict__ out) {            // [64]
    __shared__ float s_xpn[BB];
    __shared__ float s_keyn[MEM];
    __shared__ float s_mem[BB];
    __shared__ float s_mlp[BB];
    __shared__ int   s_best[BB];

    const int tid  = threadIdx.x;
    const int wave = tid >> 5;   // 0..31
    const int lane = tid & 31;

    // --- xp row norm (row = wave) ---
    {
        const float* row = xp + (size_t)wave * DD;
        float s = 0.f;
        for (int i = lane; i < DD; i += 32) { float v = row[i]; s += v * v; }
        s = wave_sum(s);
        if (lane == 0) s_xpn[wave] = sqrtf(s);
    }
    // --- key row norms (2 rows per wave) ---
    for (int r = wave; r < MEM; r += BB) {
        const float* row = key_matrix + (size_t)r * DD;
        float s = 0.f;
        for (int i = lane; i < DD; i += 32) { float v = row[i]; s += v * v; }
        s = wave_sum(s);
        if (lane == 0) s_keyn[r] = sqrtf(s);
    }
    // --- cosine sims vs mkey0/1, argmax (strict >, first-on-tie) ---
    {
        const float* row = xp + (size_t)wave * DD;
        float d0 = 0.f, d1 = 0.f, n0 = 0.f, n1 = 0.f;
        for (int i = lane; i < DD; i += 32) {
            float v = row[i], k0 = mkey0[i], k1 = mkey1[i];
            d0 += v * k0; d1 += v * k1; n0 += k0 * k0; n1 += k1 * k1;
        }
        d0 = wave_sum(d0); d1 = wave_sum(d1);
        n0 = wave_sum(n0); n1 = wave_sum(n1);
        if (lane == 0) {
            const float nx   = s_xpn[wave];
            const float sim0 = d0 / fmaxf(nx * sqrtf(n0), 1e-8f);
            const float sim1 = d1 / fmaxf(nx * sqrtf(n1), 1e-8f);
            s_best[wave] = (sim1 > sim0) ? 1 : 0;
        }
    }
    // --- mlp_out (row = wave) ---
    {
        const float* row = h + (size_t)wave * HH;
        float s = 0.f;
        for (int i = lane; i < HH; i += 32) s += row[i] * mlp_w2[i];
        s = wave_sum(s);
        if (lane == 0) s_mlp[wave] = s + mlp_b2[0];
    }
    __syncthreads();   // s_keyn, s_xpn complete

    // --- softmax over MEM=64 logits (b = wave; 2 logits per lane) ---
    {
        const float nx = fmaxf(s_xpn[wave], 1e-12f);
        const float inv_sqrtD = 1.0f / 32.0f;   // 1/sqrt(1024)
        const float* drow = dotk + (size_t)wave * MEM;
        float l0 = drow[lane]      / (nx * fmaxf(s_keyn[lane],      1e-12f)) * inv_sqrtD;
        float l1 = drow[lane + 32] / (nx * fmaxf(s_keyn[lane + 32], 1e-12f)) * inv_sqrtD;
        float mx = wave_max(fmaxf(l0, l1));
        float e0 = __expf(l0 - mx), e1 = __expf(l1 - mx);
        float den = wave_sum(e0 + e1);
        float num = wave_sum(e0 * val_matrix[lane] + e1 * val_matrix[lane + 32]);
        if (lane == 0) s_mem[wave] = num / den;
    }
    __syncthreads();

    // --- outputs: out[0:32] = scalar, out[32:64] = gate ---
    if (tid < BB) {
        const int b = tid;
        const int changed = (b == 0) ? 1 : (s_best[b] != s_best[b - 1] ? 1 : 0);
        out[b] = ((float)changed > threshold[0]) ? 1.0f : 0.0f;
        out[BB + b] = concat_w[0] * s_mem[b] + concat_w[1] * s_mlp[b] + concat_b[0];
    }
}

// ---------------------------------------------------------------------------
// Launch
// ---------------------------------------------------------------------------
extern "C" void kernel_launch(void* const* d_in, const int* in_sizes, int n_in,
                              void* d_out, int out_size, void* d_ws, size_t ws_size,
                              hipStream_t stream) {
    const float* x          = (const float*)d_in[0];
    const float* mkey0      = (const float*)d_in[1];
    const float* mkey1      = (const float*)d_in[2];
    const float* key_matrix = (const float*)d_in[3];
    const float* val_matrix = (const float*)d_in[4];
    const float* mlp_w1     = (const float*)d_in[5];
    const float* mlp_b1     = (const float*)d_in[6];
    const float* mlp_w2     = (const float*)d_in[7];
    const float* mlp_b2     = (const float*)d_in[8];
    const float* concat_w   = (const float*)d_in[9];
    const float* concat_b   = (const float*)d_in[10];
    const float* xa_w       = (const float*)d_in[11];
    const float* xa_b       = (const float*)d_in[12];
    const float* threshold  = (const float*)d_in[13];
    float* out = (float*)d_out;

    // workspace layout (floats)
    float* ws    = (float*)d_ws;
    float* part  = ws;                                   // 32*16*1024 = 524288
    float* mean  = part + (size_t)BB * SPLIT * DD;       // 32768
    float* xp    = mean + (size_t)BB * DD;               // 32768
    float* h     = xp   + (size_t)BB * DD;               // 16384
    float* dotk  = h    + (size_t)BB * HH;               // 2048

    // 1) mean over T (bandwidth-dominant: 256 MiB)
    mean_partial_kernel<<<dim3(BB, SPLIT), 256, 0, stream>>>(x, part);
    mean_final_kernel<<<BB, 256, 0, stream>>>(part, mean);

    // 2) xp = mean @ xa_w^T + xa_b        (M=32,N=1024,K=1024) — f32 WMMA
    wmma_gemm_f32_kernel<<<dim3(BB / 16, DD / 16), 32, 0, stream>>>(
        mean, xa_w, xa_b, xp, DD, DD, 0);

    // 3) h = silu(xp @ mlp_w1^T + b1)     (M=32,N=512,K=1024) — f32 WMMA
    wmma_gemm_f32_kernel<<<dim3(BB / 16, HH / 16), 32, 0, stream>>>(
        xp, mlp_w1, mlp_b1, h, HH, DD, 1);

    // 4) dotk = xp @ key_matrix^T         (M=32,N=64,K=1024) — f32 WMMA
    wmma_gemm_f32_kernel<<<dim3(BB / 16, MEM / 16), 32, 0, stream>>>(
        xp, key_matrix, nullptr, dotk, MEM, DD, 0);

    // 5) fused norms / sims / softmax / gate -> d_out[0:64]
    finalize_kernel<<<1, 1024, 0, stream>>>(
        xp, dotk, h, mkey0, mkey1, val_matrix, mlp_w2, mlp_b2,
        concat_w, concat_b, key_matrix, threshold, out);
}